// RGCNCell_49624052138542
// MI455X (gfx1250) — compile-verified
//
#include <hip/hip_runtime.h>

#define RANK 128
#define NE 50000
#define NEDGE 500000
#define SLOPE 0.22916666666666666f  // (1/8 + 1/3)/2

typedef __attribute__((ext_vector_type(2))) float v2f;
typedef __attribute__((ext_vector_type(8))) float v8f;

// ---------- zero helpers ----------
__global__ void zero_f4(float4* __restrict__ p, int n4) {
    int i = blockIdx.x * blockDim.x + threadIdx.x;
    if (i < n4) p[i] = make_float4(0.f, 0.f, 0.f, 0.f);
}
__global__ void zero_i32(int* __restrict__ p, int n) {
    int i = blockIdx.x * blockDim.x + threadIdx.x;
    if (i < n) p[i] = 0;
}

// ---------- weight transpose (once per launch; 128x128, L2-resident) ----------
// WT[n][k] = W[k][n]  -> B fragments become contiguous b64 loads in the GEMM.
__global__ void transpose_w(const float* __restrict__ W, float* __restrict__ WT) {
    int i = blockIdx.x * blockDim.x + threadIdx.x;   // 0 .. 128*128-1
    int r = i >> 7, c = i & 127;
    WT[(size_t)c * RANK + r] = W[i];
}

// ---------- in-degree (once; graph identical across layers) ----------
__global__ void deg_kernel(const int* __restrict__ dst, int* __restrict__ deg, int n) {
    int i = blockIdx.x * blockDim.x + threadIdx.x;
    if (i < n) atomicAdd(&deg[dst[i]], 1);
}

// ---------- edge phase: pre[dst] += h[src] + rel[etype] (pure data movement) ----------
// One wave per edge iteration: 32 lanes x float4 = 128 floats.
__global__ void edge_agg(const float* __restrict__ h, const float* __restrict__ rel,
                         const int* __restrict__ src, const int* __restrict__ dst,
                         const int* __restrict__ ety, float* __restrict__ pre, int n) {
    const int wid    = (blockIdx.x * blockDim.x + threadIdx.x) >> 5;
    const int lane   = threadIdx.x & 31;
    const int nwaves = (gridDim.x * blockDim.x) >> 5;
    for (int e = wid; e < n; e += nwaves) {
        const int s = src[e], d = dst[e], r = ety[e];
        const float4 hv = *(const float4*)(h   + (size_t)s * RANK + lane * 4);
        const float4 rv = *(const float4*)(rel + (size_t)r * RANK + lane * 4);
        float* p = pre + (size_t)d * RANK + lane * 4;
        __hip_atomic_fetch_add(p + 0, hv.x + rv.x, __ATOMIC_RELAXED, __HIP_MEMORY_SCOPE_AGENT);
        __hip_atomic_fetch_add(p + 1, hv.y + rv.y, __ATOMIC_RELAXED, __HIP_MEMORY_SCOPE_AGENT);
        __hip_atomic_fetch_add(p + 2, hv.z + rv.z, __ATOMIC_RELAXED, __HIP_MEMORY_SCOPE_AGENT);
        __hip_atomic_fetch_add(p + 3, hv.w + rv.w, __ATOMIC_RELAXED, __HIP_MEMORY_SCOPE_AGENT);
    }
}

// ---------- node phase: out = lrelu(pre@Wn * norm + h@(deg>0 ? Wl : Wa)) ----------
// One wave owns a 16x16 output block; 3 WMMA f32 16x16x4 accumulator chains.
// Weights are pre-transposed: B fragment = WT[n0+l16][k+half*2 .. +1] (b64 load).
// Row tiles: 50000/16 = 3125 (exact). Col tiles: 128/16 = 8. 25000 waves total.
__global__ void __launch_bounds__(256) node_update(
    const float* __restrict__ pre, const float* __restrict__ h,
    const float* __restrict__ WnT, const float* __restrict__ WlT, const float* __restrict__ WaT,
    const float* __restrict__ norm, const int* __restrict__ deg, float* __restrict__ out) {
    const int wid  = (blockIdx.x * blockDim.x + threadIdx.x) >> 5;
    const int lane = threadIdx.x & 31;
    const int row0 = (wid >> 3) * 16;
    const int n0   = (wid & 7) * 16;
    const int half = lane >> 4;   // 0: K+0/K+1, 1: K+2/K+3 (ISA f32 A/B fragment layout)
    const int l16  = lane & 15;

    const float* preRow = pre + (size_t)(row0 + l16) * RANK + half * 2;
    const float* hRow   = h   + (size_t)(row0 + l16) * RANK + half * 2;
    const float* WnC = WnT + (size_t)(n0 + l16) * RANK + half * 2;
    const float* WlC = WlT + (size_t)(n0 + l16) * RANK + half * 2;
    const float* WaC = WaT + (size_t)(n0 + l16) * RANK + half * 2;

    v8f cN = {}, cL = {}, cA = {};
    #pragma unroll 4
    for (int k = 0; k < RANK; k += 4) {
        v2f aP = *(const v2f*)(preRow + k);   // 8B aligned -> global_load_b64
        v2f aH = *(const v2f*)(hRow + k);
        v2f bN = *(const v2f*)(WnC + k);
        v2f bL = *(const v2f*)(WlC + k);
        v2f bA = *(const v2f*)(WaC + k);
        cN = __builtin_amdgcn_wmma_f32_16x16x4_f32(false, aP, false, bN, (short)0, cN, false, false);
        cL = __builtin_amdgcn_wmma_f32_16x16x4_f32(false, aH, false, bL, (short)0, cL, false, false);
        cA = __builtin_amdgcn_wmma_f32_16x16x4_f32(false, aH, false, bA, (short)0, cA, false, false);
    }

    const int col = n0 + l16;
    #pragma unroll
    for (int j = 0; j < 8; ++j) {
        const int m = row0 + j + half * 8;   // C layout: VGPR j -> M=j (lo lanes) / M=j+8 (hi lanes)
        const float nrm = norm[m];
        const float lp  = (deg[m] > 0) ? cL[j] : cA[j];
        float v = cN[j] * nrm + lp;
        v = (v >= 0.f) ? v : v * SLOPE;
        out[(size_t)m * RANK + col] = v;
    }
}

extern "C" void kernel_launch(void* const* d_in, const int* in_sizes, int n_in,
                              void* d_out, int out_size, void* d_ws, size_t ws_size,
                              hipStream_t stream) {
    const float* ent  = (const float*)d_in[0];
    const float* rel  = (const float*)d_in[1];
    const float* norm = (const float*)d_in[2];
    const float* Wn   = (const float*)d_in[3];   // [2,128,128]
    const float* Wl   = (const float*)d_in[4];
    const float* Wa   = (const float*)d_in[5];
    const int*   src  = (const int*)d_in[6];
    const int*   dst  = (const int*)d_in[7];
    const int*   ety  = (const int*)d_in[8];
    float* out = (float*)d_out;

    char* ws = (char*)d_ws;
    const size_t HB = (size_t)NE * RANK * sizeof(float);   // 25.6 MB
    const size_t WB = (size_t)RANK * RANK * sizeof(float); // 64 KB
    float* pre = (float*)(ws);
    float* h1  = (float*)(ws + HB);
    int*   deg = (int*)(ws + 2 * HB);
    float* WT  = (float*)(ws + 2 * HB + ((size_t)NE * sizeof(int) + 255 & ~(size_t)255));
    // WT layout: [WnT0, WlT0, WaT0, WnT1, WlT1, WaT1], each 128x128
    float* WnT0 = WT + 0 * RANK * RANK;
    float* WlT0 = WT + 1 * RANK * RANK;
    float* WaT0 = WT + 2 * RANK * RANK;
    float* WnT1 = WT + 3 * RANK * RANK;
    float* WlT1 = WT + 4 * RANK * RANK;
    float* WaT1 = WT + 5 * RANK * RANK;

    const int n4 = NE * RANK / 4;                 // 1.6M float4
    const int zb = (n4 + 255) / 256;              // 6250 blocks
    const int nodeBlocks = NE / 16;               // 3125 blocks x 8 waves = 25000 tiles
    const int tb = RANK * RANK / 256;             // 64 blocks per transpose
    const size_t WOFF = (size_t)RANK * RANK;

    // in-degree (once) + weight transposes
    zero_i32<<<(NE + 255) / 256, 256, 0, stream>>>(deg, NE);
    deg_kernel<<<(NEDGE + 255) / 256, 256, 0, stream>>>(dst, deg, NEDGE);
    transpose_w<<<tb, 256, 0, stream>>>(Wn,        WnT0);
    transpose_w<<<tb, 256, 0, stream>>>(Wl,        WlT0);
    transpose_w<<<tb, 256, 0, stream>>>(Wa,        WaT0);
    transpose_w<<<tb, 256, 0, stream>>>(Wn + WOFF, WnT1);
    transpose_w<<<tb, 256, 0, stream>>>(Wl + WOFF, WlT1);
    transpose_w<<<tb, 256, 0, stream>>>(Wa + WOFF, WaT1);

    // ---- layer 0 ----
    zero_f4<<<zb, 256, 0, stream>>>((float4*)pre, n4);
    edge_agg<<<2048, 256, 0, stream>>>(ent, rel, src, dst, ety, pre, NEDGE);
    node_update<<<nodeBlocks, 256, 0, stream>>>(pre, ent, WnT0, WlT0, WaT0, norm, deg, h1);

    // ---- layer 1 ----
    zero_f4<<<zb, 256, 0, stream>>>((float4*)pre, n4);
    edge_agg<<<2048, 256, 0, stream>>>(h1, rel, src, dst, ety, pre, NEDGE);
    node_update<<<nodeBlocks, 256, 0, stream>>>(pre, h1, WnT1, WlT1, WaT1, norm, deg, out);
}